// TFN_48902497632454
// MI455X (gfx1250) — compile-verified
//
#include <hip/hip_runtime.h>
#include <math.h>

// ---------------------------------------------------------------------------
// SE(3)-equivariant TFN forward for MI455X (gfx1250, wave32).
// Radial MLPs run on V_WMMA_F32_16X16X4_F32 in TRANSPOSED form:
//   D = W * X^T  (A = weight rows, B = 16-edge activation tile)
// so each lane owns one edge; LayerNorm needs one v_permlanex16_b32 and the
// message contractions are in-lane dots accumulated via ds_add_f32.
// ---------------------------------------------------------------------------

#define N_NODES 10000
#define N_EDGES 40000
#define EPS_R   1e-8f

typedef __attribute__((ext_vector_type(2))) float v2f;
typedef __attribute__((ext_vector_type(8))) float v8f;

// Radial-net parameter pointers, in sorted-key flatten order:
// b1,b2,b3,bb1,bb2,g1,g2,w1,w2,w3   (w1 is replaced by a [32][8] padded copy)
struct RP {
  const float *b1, *b2, *b3, *bb1, *bb2, *g1, *g2, *w1, *w2, *w3;
};

__device__ __forceinline__ v8f wmma4(v2f a, v2f b, v8f c) {
  return __builtin_amdgcn_wmma_f32_16x16x4_f32(
      /*neg_a=*/false, a, /*neg_b=*/false, b,
      /*c_mod=*/(short)0, c, /*reuse_a=*/false, /*reuse_b=*/false);
}

// Exchange with lane^16 (identity selector crossing the two 16-lane rows).
__device__ __forceinline__ float permx16(float x) {
  int xi = __float_as_int(x);
  int r = __builtin_amdgcn_permlanex16(xi, xi, 0x76543210, (int)0xFEDCBA98,
                                       /*fi=*/false, /*bound_ctrl=*/false);
  return __int_as_float(r);
}

__device__ __forceinline__ v2f ldv2(const float* p) {
  return *reinterpret_cast<const v2f*>(p);
}
__device__ __forceinline__ float4 ld4(const float* p) {
  return *reinterpret_cast<const float4*>(p);
}

// In-lane dot of a D-tile column (8 channels) with gathered source features.
__device__ __forceinline__ float dot8(const v8f& a, const float* s) {
  float p = 0.0f;
#pragma unroll
  for (int r = 0; r < 8; ++r) p = fmaf(a[r], s[r], p);
  return p;
}

// LayerNorm(32)+ReLU on transposed tiles: lane holds channels rowbase+r (t0)
// and 16+rowbase+r (t1) of edge (lane&15); partner channels are in lane^16.
__device__ __forceinline__ void lnT(v8f& t0, v8f& t1, const float* g,
                                    const float* bb, int rowbase) {
  float s = 0.0f, s2 = 0.0f;
#pragma unroll
  for (int r = 0; r < 8; ++r) {
    s += t0[r] + t1[r];
    s2 = fmaf(t0[r], t0[r], s2);
    s2 = fmaf(t1[r], t1[r], s2);
  }
  s  += permx16(s);
  s2 += permx16(s2);
  float mu   = s * (1.0f / 32.0f);
  float var  = s2 * (1.0f / 32.0f) - mu * mu;
  float rstd = rsqrtf(fmaxf(var, 0.0f) + 1e-5f);
  float ga[16], ba[16];
  *reinterpret_cast<float4*>(ga + 0)  = ld4(g + rowbase);
  *reinterpret_cast<float4*>(ga + 4)  = ld4(g + rowbase + 4);
  *reinterpret_cast<float4*>(ga + 8)  = ld4(g + 16 + rowbase);
  *reinterpret_cast<float4*>(ga + 12) = ld4(g + 16 + rowbase + 4);
  *reinterpret_cast<float4*>(ba + 0)  = ld4(bb + rowbase);
  *reinterpret_cast<float4*>(ba + 4)  = ld4(bb + rowbase + 4);
  *reinterpret_cast<float4*>(ba + 8)  = ld4(bb + 16 + rowbase);
  *reinterpret_cast<float4*>(ba + 12) = ld4(bb + 16 + rowbase + 4);
#pragma unroll
  for (int r = 0; r < 8; ++r) {
    float x0 = (t0[r] - mu) * rstd * ga[r] + ba[r];
    float x1 = (t1[r] - mu) * rstd * ga[8 + r] + ba[8 + r];
    t0[r] = fmaxf(x0, 0.0f);
    t1[r] = fmaxf(x1, 0.0f);
  }
}

// Store transposed tiles to LDS as Hs[edge][channel] (stride 34).
__device__ __forceinline__ void storeT(float* lh, const v8f& t0, const v8f& t1,
                                       int e, int rowbase) {
#pragma unroll
  for (int r = 0; r < 8; r += 2) {
    v2f p0; p0.x = t0[r]; p0.y = t0[r + 1];
    v2f p1; p1.x = t1[r]; p1.y = t1[r + 1];
    *reinterpret_cast<v2f*>(lh + e * 34 + rowbase + r)      = p0;
    *reinterpret_cast<v2f*>(lh + e * 34 + 16 + rowbase + r) = p1;
  }
}

// Transposed radial MLP for a 16-edge tile.
//   lf: LDS [16][8]  padded input features
//   lh: LDS [16][34] per-wave hidden scratch (Hs[edge][ch])
//   consume(t, acc): layer-3 tile t; lane holds Out[t*16+rowbase+r][lane&15]
template <int NT, class F>
__device__ __forceinline__ void radialT(const RP& rp, const float* lf,
                                        float* lh, F&& consume) {
  const int lane = threadIdx.x & 31;
  const int e    = lane & 15;
  const int kb2  = (lane >> 4) * 2;
  const int rowbase = (lane >> 4) * 8;

  v8f t0, t1;
  // ---- layer 1: W1[32,8] * X^T[8,16] ----------------------------------
  {
    float4 b0 = ld4(rp.b1 + rowbase),      b0b = ld4(rp.b1 + rowbase + 4);
    float4 b1 = ld4(rp.b1 + 16 + rowbase), b1b = ld4(rp.b1 + 16 + rowbase + 4);
    t0[0]=b0.x; t0[1]=b0.y; t0[2]=b0.z; t0[3]=b0.w;
    t0[4]=b0b.x; t0[5]=b0b.y; t0[6]=b0b.z; t0[7]=b0b.w;
    t1[0]=b1.x; t1[1]=b1.y; t1[2]=b1.z; t1[3]=b1.w;
    t1[4]=b1b.x; t1[5]=b1b.y; t1[6]=b1b.z; t1[7]=b1b.w;
  }
#pragma unroll
  for (int ks = 0; ks < 2; ++ks) {
    int k = ks * 4 + kb2;
    v2f b  = ldv2(lf + e * 8 + k);
    v2f a0 = ldv2(rp.w1 + e * 8 + k);
    v2f a1 = ldv2(rp.w1 + (e + 16) * 8 + k);
    t0 = wmma4(a0, b, t0);
    t1 = wmma4(a1, b, t1);
  }
  lnT(t0, t1, rp.g1, rp.bb1, rowbase);
  storeT(lh, t0, t1, e, rowbase);

  // ---- layer 2: W2[32,32] * H1^T[32,16] -------------------------------
  v2f breg[8];
#pragma unroll
  for (int ks = 0; ks < 8; ++ks)
    breg[ks] = ldv2(lh + e * 34 + ks * 4 + kb2);
  {
    float4 b0 = ld4(rp.b2 + rowbase),      b0b = ld4(rp.b2 + rowbase + 4);
    float4 b1 = ld4(rp.b2 + 16 + rowbase), b1b = ld4(rp.b2 + 16 + rowbase + 4);
    t0[0]=b0.x; t0[1]=b0.y; t0[2]=b0.z; t0[3]=b0.w;
    t0[4]=b0b.x; t0[5]=b0b.y; t0[6]=b0b.z; t0[7]=b0b.w;
    t1[0]=b1.x; t1[1]=b1.y; t1[2]=b1.z; t1[3]=b1.w;
    t1[4]=b1b.x; t1[5]=b1b.y; t1[6]=b1b.z; t1[7]=b1b.w;
  }
#pragma unroll
  for (int ks = 0; ks < 8; ++ks) {
    int k = ks * 4 + kb2;
    v2f a0 = ldv2(rp.w2 + e * 32 + k);
    v2f a1 = ldv2(rp.w2 + (e + 16) * 32 + k);
    t0 = wmma4(a0, breg[ks], t0);
    t1 = wmma4(a1, breg[ks], t1);
  }
  lnT(t0, t1, rp.g2, rp.bb2, rowbase);
  storeT(lh, t0, t1, e, rowbase);

  // ---- layer 3: W3[16*NT,32] * H2^T[32,16] ----------------------------
#pragma unroll
  for (int ks = 0; ks < 8; ++ks)
    breg[ks] = ldv2(lh + e * 34 + ks * 4 + kb2);
  for (int t = 0; t < NT; ++t) {
    v8f acc;
    {
      float4 blo = ld4(rp.b3 + t * 16 + rowbase);
      float4 bhi = ld4(rp.b3 + t * 16 + rowbase + 4);
      acc[0]=blo.x; acc[1]=blo.y; acc[2]=blo.z; acc[3]=blo.w;
      acc[4]=bhi.x; acc[5]=bhi.y; acc[6]=bhi.z; acc[7]=bhi.w;
    }
    const float* w3t = rp.w3 + (t * 16 + e) * 32 + kb2;
#pragma unroll
    for (int ks = 0; ks < 8; ++ks)
      acc = wmma4(ldv2(w3t + ks * 4), breg[ks], acc);
    consume(t, acc);
  }
}

// ---------------------------------------------------------------------------
// w1 pre-pack: eight radial w1 [32,5] -> zero-padded [32,8] in workspace.
// ---------------------------------------------------------------------------
struct W1Ptrs { const float* p[8]; };

__global__ void tfn_pack_w1(W1Ptrs src, float* __restrict__ dst) {
  int i = blockIdx.x * blockDim.x + threadIdx.x;  // 0..2047
  int rid = i >> 8;
  int j = i & 255;
  int n = j >> 3, k = j & 7;
  dst[i] = (k < 5) ? src.p[rid][n * 5 + k] : 0.0f;
}

// ---------------------------------------------------------------------------
// Edge preprocessing.
// ---------------------------------------------------------------------------
__global__ void tfn_edge_prep(const int* __restrict__ ei,
                              const float* __restrict__ pos,
                              const float* __restrict__ ef,
                              float* __restrict__ featPad,
                              float* __restrict__ uvec,
                              float* __restrict__ deg) {
  int e = blockIdx.x * blockDim.x + threadIdx.x;
  if (e >= N_EDGES) return;
  int rn = ei[e], cn = ei[N_EDGES + e];
  float dx = pos[rn * 3 + 0] - pos[cn * 3 + 0];
  float dy = pos[rn * 3 + 1] - pos[cn * 3 + 1];
  float dz = pos[rn * 3 + 2] - pos[cn * 3 + 2];
  float rr = sqrtf(dx * dx + dy * dy + dz * dz + EPS_R);
  float ir = 1.0f / rr;
  *reinterpret_cast<float4*>(uvec + e * 4) =
      make_float4(dx * ir, dy * ir, dz * ir, 0.0f);
  *reinterpret_cast<float4*>(featPad + e * 8) =
      make_float4(ef[e * 4 + 0], ef[e * 4 + 1], ef[e * 4 + 2], ef[e * 4 + 3]);
  *reinterpret_cast<float4*>(featPad + e * 8 + 4) =
      make_float4(rr, 0.f, 0.f, 0.f);
  atomicAdd(&deg[rn], 1.0f);
}

// ---------------------------------------------------------------------------
// conv0: (deg0,16) -> {(deg0,16),(deg1,16)}  pairs r00, r01
// ---------------------------------------------------------------------------
__global__ __launch_bounds__(128) void tfn_conv0(
    const int* __restrict__ ei, const float* __restrict__ feat,
    const float* __restrict__ uvec, const float* __restrict__ h0,
    float* __restrict__ agg0, float* __restrict__ agg1, RP r00, RP r01) {
  __shared__ float sFeat[4][16 * 8];
  __shared__ float sH[4][16 * 34];
  __shared__ float sS0[4][16 * 16];
  __shared__ float sM0[4][256];
  __shared__ float sT01[4][256];
  const int w = threadIdx.x >> 5, lane = threadIdx.x & 31;
  const int e0 = (blockIdx.x * 4 + w) * 16;
  float* lf   = sFeat[w];
  float* lh   = sH[w];
  float* ls0  = sS0[w];
  float* lm0  = sM0[w];
  float* lt01 = sT01[w];

  reinterpret_cast<float4*>(lf)[lane] =
      reinterpret_cast<const float4*>(feat + e0 * 8)[lane];
  for (int p = lane; p < 256; p += 32) {
    int m = p >> 4, c = p & 15;
    int src = ei[N_EDGES + e0 + m];
    ls0[p] = h0[src * 16 + c];
    lm0[p] = 0.0f;
    lt01[p] = 0.0f;
  }
  const int e = lane & 15, rowbase = (lane >> 4) * 8;
  float s0p[8];
#pragma unroll
  for (int r = 0; r < 8; ++r) s0p[r] = ls0[e * 16 + rowbase + r];

  radialT<16>(r00, lf, lh, [&](int t, const v8f& acc) {
    atomicAdd(&lm0[t * 16 + e], dot8(acc, s0p));
  });
  radialT<16>(r01, lf, lh, [&](int t, const v8f& acc) {
    atomicAdd(&lt01[t * 16 + e], dot8(acc, s0p));
  });

  for (int i = lane; i < 256; i += 32) {
    int mo = i >> 4, ee = i & 15;
    int dst = ei[e0 + ee];
    atomicAdd(&agg0[dst * 32 + mo], lm0[i]);
    float tv = lt01[i];
    float4 u4 = ld4(uvec + (e0 + ee) * 4);
    float* a1 = &agg1[(dst * 16 + mo) * 3];
    atomicAdd(a1 + 0, tv * u4.x);
    atomicAdd(a1 + 1, tv * u4.y);
    atomicAdd(a1 + 2, tv * u4.z);
  }
}

// ---------------------------------------------------------------------------
// conv1: {0,1} -> {0,1}   pairs r00, r01, r10, r11   (2 waves per block)
// ---------------------------------------------------------------------------
__global__ __launch_bounds__(64) void tfn_conv1(
    const int* __restrict__ ei, const float* __restrict__ feat,
    const float* __restrict__ uvec, const float* __restrict__ h0,
    const float* __restrict__ h1, float* __restrict__ agg0,
    float* __restrict__ agg1, RP r00, RP r01, RP r10, RP r11) {
  __shared__ float sFeat[2][16 * 8];
  __shared__ float sH[2][16 * 34];
  __shared__ float sS0[2][16 * 16];
  __shared__ float sS10[2][16 * 16];
  __shared__ float sG[2][16 * 192];   // G[e][q=mi*3+f][4] padded float4
  __shared__ float sM0[2][256];
  __shared__ float sT01[2][256];
  __shared__ float sM1[2][256 * 4];   // msg1[(mo*16+e)*4 + j]
  const int w = threadIdx.x >> 5, lane = threadIdx.x & 31;
  const int e0 = (blockIdx.x * 2 + w) * 16;
  float* lf   = sFeat[w];
  float* lh   = sH[w];
  float* ls0  = sS0[w];
  float* ls10 = sS10[w];
  float* lg   = sG[w];
  float* lm0  = sM0[w];
  float* lt01 = sT01[w];
  float* lm1  = sM1[w];

  const float IS3 = 0.5773502691896258f;  // 1/sqrt(3)
  const float IS2 = 0.7071067811865476f;  // 1/sqrt(2)

  reinterpret_cast<float4*>(lf)[lane] =
      reinterpret_cast<const float4*>(feat + e0 * 8)[lane];
  for (int p = lane; p < 256; p += 32) {
    int m = p >> 4, c = p & 15;
    int ee = e0 + m;
    int src = ei[N_EDGES + ee];
    ls0[p] = h0[src * 16 + c];
    float hx = h1[(src * 16 + c) * 3 + 0];
    float hy = h1[(src * 16 + c) * 3 + 1];
    float hz = h1[(src * 16 + c) * 3 + 2];
    float4 u4 = ld4(uvec + ee * 4);
    float hu = hx * u4.x + hy * u4.y + hz * u4.z;
    ls10[p] = hu;
    float* Gp = &lg[m * 192 + c * 12];   // q = c*3+f, stride 4
    *reinterpret_cast<float4*>(Gp + 0) =
        make_float4(hx * IS3, hy * IS3, hz * IS3, 0.f);
    *reinterpret_cast<float4*>(Gp + 4) =
        make_float4((hy * u4.z - hz * u4.y) * IS2,
                    (hz * u4.x - hx * u4.z) * IS2,
                    (hx * u4.y - hy * u4.x) * IS2, 0.f);
    *reinterpret_cast<float4*>(Gp + 8) =
        make_float4(u4.x * hu - hx * (1.0f / 3.0f),
                    u4.y * hu - hy * (1.0f / 3.0f),
                    u4.z * hu - hz * (1.0f / 3.0f), 0.f);
    lm0[p] = 0.0f;
    lt01[p] = 0.0f;
    float4 z4 = make_float4(0.f, 0.f, 0.f, 0.f);
    *reinterpret_cast<float4*>(lm1 + p * 4) = z4;
  }
  const int e = lane & 15, rowbase = (lane >> 4) * 8;
  float s0p[8], s10p[8];
#pragma unroll
  for (int r = 0; r < 8; ++r) {
    s0p[r]  = ls0[e * 16 + rowbase + r];
    s10p[r] = ls10[e * 16 + rowbase + r];
  }

  // ---- degree-0 output: r00 + r10 --------------------------------------
  radialT<16>(r00, lf, lh, [&](int t, const v8f& acc) {
    atomicAdd(&lm0[t * 16 + e], dot8(acc, s0p));
  });
  radialT<16>(r10, lf, lh, [&](int t, const v8f& acc) {
    atomicAdd(&lm0[t * 16 + e], dot8(acc, s10p));
  });

  // ---- degree-1 output: r01 + r11 --------------------------------------
  radialT<16>(r01, lf, lh, [&](int t, const v8f& acc) {
    atomicAdd(&lt01[t * 16 + e], dot8(acc, s0p));
  });
  radialT<48>(r11, lf, lh, [&](int t, const v8f& acc) {
    int ph = t % 3, mo = t / 3;
    int qb = ph * 16;
    float p0 = 0.f, p1 = 0.f, p2 = 0.f;
#pragma unroll
    for (int r = 0; r < 8; ++r) {
      float4 Gv = ld4(lg + e * 192 + (qb + rowbase + r) * 4);
      p0 = fmaf(acc[r], Gv.x, p0);
      p1 = fmaf(acc[r], Gv.y, p1);
      p2 = fmaf(acc[r], Gv.z, p2);
    }
    float* m = &lm1[(mo * 16 + e) * 4];
    atomicAdd(m + 0, p0);
    atomicAdd(m + 1, p1);
    atomicAdd(m + 2, p2);
  });

  for (int i = lane; i < 256; i += 32) {
    int mo = i >> 4, ee = i & 15;
    int dst = ei[e0 + ee];
    atomicAdd(&agg0[dst * 32 + mo], lm0[i]);
    float tv = lt01[i];
    float4 u4 = ld4(uvec + (e0 + ee) * 4);
    float* a1 = &agg1[(dst * 16 + mo) * 3];
    atomicAdd(a1 + 0, lm1[i * 4 + 0] + tv * u4.x);
    atomicAdd(a1 + 1, lm1[i * 4 + 1] + tv * u4.y);
    atomicAdd(a1 + 2, lm1[i * 4 + 2] + tv * u4.z);
  }
}

// ---------------------------------------------------------------------------
// conv2: {0,1} -> (deg0,32)   pairs r00, r10
// ---------------------------------------------------------------------------
__global__ __launch_bounds__(128) void tfn_conv2(
    const int* __restrict__ ei, const float* __restrict__ feat,
    const float* __restrict__ uvec, const float* __restrict__ h0,
    const float* __restrict__ h1, float* __restrict__ agg0, RP r00, RP r10) {
  __shared__ float sFeat[4][16 * 8];
  __shared__ float sH[4][16 * 34];
  __shared__ float sS0[4][16 * 16];
  __shared__ float sS10[4][16 * 16];
  __shared__ float sM0[4][512];
  const int w = threadIdx.x >> 5, lane = threadIdx.x & 31;
  const int e0 = (blockIdx.x * 4 + w) * 16;
  float* lf   = sFeat[w];
  float* lh   = sH[w];
  float* ls0  = sS0[w];
  float* ls10 = sS10[w];
  float* lm0  = sM0[w];

  reinterpret_cast<float4*>(lf)[lane] =
      reinterpret_cast<const float4*>(feat + e0 * 8)[lane];
  for (int p = lane; p < 256; p += 32) {
    int m = p >> 4, c = p & 15;
    int ee = e0 + m;
    int src = ei[N_EDGES + ee];
    ls0[p] = h0[src * 16 + c];
    float hx = h1[(src * 16 + c) * 3 + 0];
    float hy = h1[(src * 16 + c) * 3 + 1];
    float hz = h1[(src * 16 + c) * 3 + 2];
    float4 u4 = ld4(uvec + ee * 4);
    ls10[p] = hx * u4.x + hy * u4.y + hz * u4.z;
    lm0[p] = 0.0f;
    lm0[p + 256] = 0.0f;
  }
  const int e = lane & 15, rowbase = (lane >> 4) * 8;
  float s0p[8], s10p[8];
#pragma unroll
  for (int r = 0; r < 8; ++r) {
    s0p[r]  = ls0[e * 16 + rowbase + r];
    s10p[r] = ls10[e * 16 + rowbase + r];
  }

  radialT<32>(r00, lf, lh, [&](int t, const v8f& acc) {
    atomicAdd(&lm0[t * 16 + e], dot8(acc, s0p));
  });
  radialT<32>(r10, lf, lh, [&](int t, const v8f& acc) {
    atomicAdd(&lm0[t * 16 + e], dot8(acc, s10p));
  });

  for (int i = lane; i < 512; i += 32) {
    int mo = i >> 4, ee = i & 15;
    int dst = ei[e0 + ee];
    atomicAdd(&agg0[dst * 32 + mo], lm0[i]);
  }
}

// ---------------------------------------------------------------------------
// Node post-processing (inv-degree, self-interaction, gated norm).
// ---------------------------------------------------------------------------
__device__ __forceinline__ void ln16_relu(const float* x, const float* g,
                                          const float* b, float* out) {
  float mu = 0.0f;
#pragma unroll
  for (int c = 0; c < 16; ++c) mu += x[c];
  mu *= (1.0f / 16.0f);
  float var = 0.0f;
#pragma unroll
  for (int c = 0; c < 16; ++c) { float d = x[c] - mu; var += d * d; }
  var *= (1.0f / 16.0f);
  float rstd = rsqrtf(var + 1e-5f);
#pragma unroll
  for (int c = 0; c < 16; ++c)
    out[c] = fmaxf((x[c] - mu) * rstd * g[c] + b[c], 0.0f);
}

__global__ void tfn_post01(const float* __restrict__ agg0,
                           const float* __restrict__ agg1,
                           const float* __restrict__ deg,
                           const float* __restrict__ h0in,
                           const float* __restrict__ h1in,
                           const float* __restrict__ self0,
                           const float* __restrict__ self1,
                           const float* __restrict__ n0g, const float* __restrict__ n0b,
                           const float* __restrict__ n0w, const float* __restrict__ n0bias,
                           const float* __restrict__ n1g, const float* __restrict__ n1b,
                           const float* __restrict__ n1w, const float* __restrict__ n1bias,
                           float* __restrict__ h0out, float* __restrict__ h1out) {
  int n = blockIdx.x * blockDim.x + threadIdx.x;
  if (n >= N_NODES) return;
  float inv = 1.0f / fmaxf(deg[n], 1.0f);

  float o0[16], o1[16][3];
  for (int mo = 0; mo < 16; ++mo) {
    float a = agg0[n * 32 + mo] * inv;
    float s = 0.0f;
    for (int i = 0; i < 16; ++i) s += self0[mo * 16 + i] * h0in[n * 16 + i];
    o0[mo] = a + s;
  }
  for (int mo = 0; mo < 16; ++mo) {
    for (int j = 0; j < 3; ++j) {
      float a = agg1[(n * 16 + mo) * 3 + j] * inv;
      if (self1 != nullptr) {
        float s = 0.0f;
        for (int i = 0; i < 16; ++i)
          s += self1[mo * 16 + i] * h1in[(n * 16 + i) * 3 + j];
        a += s;
      }
      o1[mo][j] = a;
    }
  }
  // gated norm, degree 0
  float nrm[16], t[16];
  for (int c = 0; c < 16; ++c) nrm[c] = sqrtf(o0[c] * o0[c] + EPS_R);
  ln16_relu(nrm, n0g, n0b, t);
  for (int mo = 0; mo < 16; ++mo) {
    float acc = n0bias[mo];
    for (int c = 0; c < 16; ++c) acc += n0w[mo * 16 + c] * t[c];
    h0out[n * 16 + mo] = acc * (o0[mo] / nrm[mo]);
  }
  // gated norm, degree 1
  for (int c = 0; c < 16; ++c)
    nrm[c] = sqrtf(o1[c][0] * o1[c][0] + o1[c][1] * o1[c][1] +
                   o1[c][2] * o1[c][2] + EPS_R);
  ln16_relu(nrm, n1g, n1b, t);
  for (int mo = 0; mo < 16; ++mo) {
    float acc = n1bias[mo];
    for (int c = 0; c < 16; ++c) acc += n1w[mo * 16 + c] * t[c];
    float sc = acc / nrm[mo];
    h1out[(n * 16 + mo) * 3 + 0] = sc * o1[mo][0];
    h1out[(n * 16 + mo) * 3 + 1] = sc * o1[mo][1];
    h1out[(n * 16 + mo) * 3 + 2] = sc * o1[mo][2];
  }
}

__global__ void tfn_post2(const float* __restrict__ agg0,
                          const float* __restrict__ deg,
                          const float* __restrict__ h0in,
                          const float* __restrict__ self0,
                          float* __restrict__ out) {
  int n = blockIdx.x * blockDim.x + threadIdx.x;
  if (n >= N_NODES) return;
  float inv = 1.0f / fmaxf(deg[n], 1.0f);
  for (int mo = 0; mo < 32; ++mo) {
    float s = 0.0f;
    for (int i = 0; i < 16; ++i) s += self0[mo * 16 + i] * h0in[n * 16 + i];
    out[n * 32 + mo] = agg0[n * 32 + mo] * inv + s;
  }
}

// ---------------------------------------------------------------------------
// Host launcher.
//
// Input flattening convention assumed:
//   d_in[0..3] = x, positions, edge_index, edge_feat (insertion order),
//   d_in[4..]  = params leaves, dict keys sorted at every level (JAX order):
//     conv0: r00@4 r01@14 self0@24
//     conv1: r00@25 r01@35 r10@45 r11@55 self0@65 self1@66
//     conv2: r00@67 r10@77 self0@87
//     norm0: d0{b@88 bias@89 g@90 w@91} d1{b@92 bias@93 g@94 w@95}
//     norm1: d0{96..99} d1{100..103}
//   radial dict sorted keys: b1,b2,b3,bb1,bb2,g1,g2,w1,w2,w3
// ---------------------------------------------------------------------------
static RP mkRP(void* const* d_in, int base) {
  RP r;
  r.b1  = (const float*)d_in[base + 0];
  r.b2  = (const float*)d_in[base + 1];
  r.b3  = (const float*)d_in[base + 2];
  r.bb1 = (const float*)d_in[base + 3];
  r.bb2 = (const float*)d_in[base + 4];
  r.g1  = (const float*)d_in[base + 5];
  r.g2  = (const float*)d_in[base + 6];
  r.w1  = (const float*)d_in[base + 7];
  r.w2  = (const float*)d_in[base + 8];
  r.w3  = (const float*)d_in[base + 9];
  return r;
}

extern "C" void kernel_launch(void* const* d_in, const int* in_sizes, int n_in,
                              void* d_out, int out_size, void* d_ws,
                              size_t ws_size, hipStream_t stream) {
  (void)in_sizes; (void)n_in; (void)out_size; (void)ws_size;

  const float* x   = (const float*)d_in[0];
  const float* pos = (const float*)d_in[1];
  const int*   ei  = (const int*)d_in[2];
  const float* ef  = (const float*)d_in[3];
  auto F = [&](int i) { return (const float*)d_in[i]; };

  RP c0r00 = mkRP(d_in, 4),  c0r01 = mkRP(d_in, 14);
  const float* c0self0 = F(24);
  RP c1r00 = mkRP(d_in, 25), c1r01 = mkRP(d_in, 35);
  RP c1r10 = mkRP(d_in, 45), c1r11 = mkRP(d_in, 55);
  const float* c1self0 = F(65);
  const float* c1self1 = F(66);
  RP c2r00 = mkRP(d_in, 67), c2r10 = mkRP(d_in, 77);
  const float* c2self0 = F(87);

  // Workspace layout (floats)
  float* ws      = (float*)d_ws;
  float* featPad = ws + 0;                         // E*8
  float* uvec    = ws + (size_t)N_EDGES * 8;       // E*4
  float* deg     = uvec + (size_t)N_EDGES * 4;     // N
  float* h0a     = deg + N_NODES;                  // N*16
  float* h1a     = h0a + (size_t)N_NODES * 16;     // N*48
  float* h0b     = h1a + (size_t)N_NODES * 48;     // N*16
  float* h1b     = h0b + (size_t)N_NODES * 16;     // N*48
  float* agg0    = h1b + (size_t)N_NODES * 48;     // N*32
  float* agg1    = agg0 + (size_t)N_NODES * 32;    // N*48
  float* w1pad   = agg1 + (size_t)N_NODES * 48;    // 8*256 padded w1

  // Pack the eight w1 matrices into zero-padded [32][8] form.
  W1Ptrs w1p;
  w1p.p[0] = c0r00.w1; w1p.p[1] = c0r01.w1;
  w1p.p[2] = c1r00.w1; w1p.p[3] = c1r01.w1;
  w1p.p[4] = c1r10.w1; w1p.p[5] = c1r11.w1;
  w1p.p[6] = c2r00.w1; w1p.p[7] = c2r10.w1;
  tfn_pack_w1<<<8, 256, 0, stream>>>(w1p, w1pad);
  c0r00.w1 = w1pad + 0 * 256; c0r01.w1 = w1pad + 1 * 256;
  c1r00.w1 = w1pad + 2 * 256; c1r01.w1 = w1pad + 3 * 256;
  c1r10.w1 = w1pad + 4 * 256; c1r11.w1 = w1pad + 5 * 256;
  c2r00.w1 = w1pad + 6 * 256; c2r10.w1 = w1pad + 7 * 256;

  const int EB4 = N_EDGES / 64;         // 4 waves x 16 edges
  const int EB2 = N_EDGES / 32;         // 2 waves x 16 edges
  const int NB  = (N_NODES + 255) / 256;

  hipMemsetAsync(deg, 0, N_NODES * sizeof(float), stream);
  tfn_edge_prep<<<(N_EDGES + 255) / 256, 256, 0, stream>>>(ei, pos, ef,
                                                           featPad, uvec, deg);

  // ---- layer 0 ----
  hipMemsetAsync(agg0, 0, (size_t)N_NODES * 32 * sizeof(float), stream);
  hipMemsetAsync(agg1, 0, (size_t)N_NODES * 48 * sizeof(float), stream);
  tfn_conv0<<<EB4, 128, 0, stream>>>(ei, featPad, uvec, x, agg0, agg1,
                                     c0r00, c0r01);
  tfn_post01<<<NB, 256, 0, stream>>>(agg0, agg1, deg, x, nullptr, c0self0,
                                     nullptr,
                                     F(90), F(88), F(91), F(89),   // norm0 d0
                                     F(94), F(92), F(95), F(93),   // norm0 d1
                                     h0a, h1a);

  // ---- layer 1 ----
  hipMemsetAsync(agg0, 0, (size_t)N_NODES * 32 * sizeof(float), stream);
  hipMemsetAsync(agg1, 0, (size_t)N_NODES * 48 * sizeof(float), stream);
  tfn_conv1<<<EB2, 64, 0, stream>>>(ei, featPad, uvec, h0a, h1a, agg0, agg1,
                                    c1r00, c1r01, c1r10, c1r11);
  tfn_post01<<<NB, 256, 0, stream>>>(agg0, agg1, deg, h0a, h1a, c1self0,
                                     c1self1,
                                     F(98), F(96), F(99), F(97),    // norm1 d0
                                     F(102), F(100), F(103), F(101),// norm1 d1
                                     h0b, h1b);

  // ---- layer 2 ----
  hipMemsetAsync(agg0, 0, (size_t)N_NODES * 32 * sizeof(float), stream);
  tfn_conv2<<<EB4, 128, 0, stream>>>(ei, featPad, uvec, h0b, h1b, agg0,
                                     c2r00, c2r10);
  tfn_post2<<<NB, 256, 0, stream>>>(agg0, deg, h0b, c2self0, (float*)d_out);
}